// myGATConv_62775241998418
// MI455X (gfx1250) — compile-verified
//
#include <hip/hip_runtime.h>
#include <math.h>

#define N_NODES  50000
#define N_EDGES  800000
#define NHEAD    8
#define D_OUT    32
#define D_IN     256
#define E_FEAT   64
#define N_ETYPES 5
#define NEG_SLOPE 0.2f
#define DHID     (NHEAD * D_OUT)   // 256

typedef float v8f __attribute__((ext_vector_type(8)));
typedef float v2f __attribute__((ext_vector_type(2)));

// ---------------------------------------------------------------- utilities
__global__ void fill_kernel(float* __restrict__ p, float v, int n) {
    int i = blockIdx.x * blockDim.x + threadIdx.x;
    if (i < n) p[i] = v;
}

// float atomic-max via int monotonicity trick (works with -inf init;
// positive floats ordered as signed ints, negative floats reverse-ordered
// as unsigned ints).
__device__ __forceinline__ void atomic_max_f32(float* addr, float val) {
    if (val >= 0.0f) atomicMax((int*)addr, __float_as_int(val));
    else             atomicMin((unsigned int*)addr, __float_as_uint(val));
}

// ---------------------------------------------------------------- 1) h = x@W
// WMMA f32 16x16x4. Block = 256 threads = 8 waves. Block owns rows
// [16*bid, 16*bid+16); wave w owns column tiles 2w and 2w+1 (8 waves * 2 * 16
// = 256 cols). Exact tiling: 50000/16 = 3125 blocks, EXEC all-1s everywhere
// (hard requirement for WMMA).
//
// W (256x256 f32 = 256 KB) is staged once per block into LDS via gfx1250
// async copies (GLOBAL_LOAD_ASYNC_TO_LDS_B128, ASYNCcnt), so the steady-state
// VMEM stream is only the 51 MB of x; B operands come from ds_load.
//
// A layout (32-bit 16x4): lanes 0-15 hold M=lane, VGPR j = K=j;
//                         lanes 16-31 hold M=lane-16, VGPR j = K=j+2.
// B layout mirrors (4x16): VGPR j, lanes 0-15 = K=j row, N=lane;
//                          lanes 16-31 = K=j+2 row.
// C/D layout: VGPR v, lanes 0-15 -> M=v, N=lane; lanes 16-31 -> M=v+8.
__global__ void gemm_h_kernel(const float* __restrict__ x,
                              const float* __restrict__ W,
                              float* __restrict__ h) {
    extern __shared__ float Wlds[];         // D_IN*DHID floats = 256 KB

    // ---- async-stage W into LDS (16B chunks, 64 per thread) ----
    {
        // low 32 bits of the generic pointer == LDS byte offset (ISA 10.2)
        unsigned lbase = (unsigned)(size_t)(void*)Wlds;
        unsigned long long gbase = (unsigned long long)(size_t)W;
        #pragma unroll 4
        for (int i = 0; i < (D_IN * DHID * 4) / 16 / 256; ++i) {  // 64 iters
            int chunk = i * 256 + (int)threadIdx.x;               // 16B chunk
            unsigned loff = lbase + (unsigned)chunk * 16u;
            unsigned long long gaddr = gbase + (unsigned long long)chunk * 16u;
            asm volatile("global_load_async_to_lds_b128 %0, %1, off"
                         :: "v"(loff), "v"(gaddr) : "memory");
        }
        asm volatile("s_wait_asynccnt 0" ::: "memory");
        __syncthreads();   // join all waves' staged chunks
    }

    const int lane = threadIdx.x & 31;
    const int wave = threadIdx.x >> 5;      // 0..7
    const int half = lane >> 4;             // 0/1 => K offset 2*half
    const int lm   = lane & 15;
    const int m0   = blockIdx.x * 16;
    const int n0a  = (2 * wave + 0) * 16;
    const int n0b  = (2 * wave + 1) * 16;

    v8f acc0 = {};
    v8f acc1 = {};
    const float* xrow = x + (size_t)(m0 + lm) * D_IN + 2 * half;

    for (int k = 0; k < D_IN; k += 4) {
        v2f a = *(const v2f*)(xrow + k);          // {A[M][k+2h], A[M][k+2h+1]}
        int kr = k + 2 * half;
        v2f b0, b1;
        b0.x = Wlds[(kr + 0) * DHID + n0a + lm];  // ds_load_b32
        b0.y = Wlds[(kr + 1) * DHID + n0a + lm];
        b1.x = Wlds[(kr + 0) * DHID + n0b + lm];
        b1.y = Wlds[(kr + 1) * DHID + n0b + lm];
        acc0 = __builtin_amdgcn_wmma_f32_16x16x4_f32(false, a, false, b0,
                                                     (short)0, acc0, false, false);
        acc1 = __builtin_amdgcn_wmma_f32_16x16x4_f32(false, a, false, b1,
                                                     (short)0, acc1, false, false);
    }

    #pragma unroll
    for (int v = 0; v < 8; ++v) {
        int r = m0 + v + 8 * half;
        float c0 = acc0[v]; c0 = (c0 != c0) ? 0.0f : c0;   // NaN -> 0
        float c1 = acc1[v]; c1 = (c1 != c1) ? 0.0f : c1;
        h[(size_t)r * DHID + n0a + lm] = c0;
        h[(size_t)r * DHID + n0b + lm] = c1;
    }
}

// ---------------------------------------------------- 2) hl/hr per node-head
__global__ void node_attn_kernel(const float* __restrict__ h,
                                 const float* __restrict__ a_l,
                                 const float* __restrict__ a_r,
                                 float* __restrict__ hl,
                                 float* __restrict__ hr) {
    int idx = blockIdx.x * blockDim.x + threadIdx.x;   // node*8 + head
    if (idx >= N_NODES * NHEAD) return;
    int head = idx & (NHEAD - 1);
    const float* hp = h + (size_t)idx * D_OUT;         // = node*256 + head*32
    float sl = 0.0f, sr = 0.0f;
    #pragma unroll
    for (int d = 0; d < D_OUT; ++d) {
        float v = hp[d];
        sl += a_l[head * D_OUT + d] * v;
        sr += a_r[head * D_OUT + d] * v;
    }
    hl[idx] = sl;
    hr[idx] = sr;
}

// ------------------------------------------- 3) he[t][hd] (tiny micro-GEMM)
__global__ void etype_attn_kernel(const float* __restrict__ edge_emb,
                                  const float* __restrict__ W_e,
                                  const float* __restrict__ a_e,
                                  float* __restrict__ he) {
    __shared__ float red[64];
    int t  = blockIdx.x >> 3;   // etype
    int hd = blockIdx.x & 7;    // head
    int f  = threadIdx.x;       // 0..63
    float s = 0.0f;
    for (int k = 0; k < E_FEAT; ++k)
        s += edge_emb[t * E_FEAT + k] *
             W_e[(size_t)k * (E_FEAT * NHEAD) + hd * E_FEAT + f];
    s *= a_e[hd * E_FEAT + f];
    red[f] = s;
    __syncthreads();
    for (int off = 32; off > 0; off >>= 1) {
        if (f < off) red[f] += red[f + off];
        __syncthreads();
    }
    if (f == 0) he[t * NHEAD + hd] = red[0];
}

// -------------------------------- 4) per-edge logits + segment max over col
__global__ void logits_kernel(const float* __restrict__ hl,
                              const float* __restrict__ hr,
                              const float* __restrict__ he,
                              const long long* __restrict__ row,
                              const long long* __restrict__ col,
                              const long long* __restrict__ etype,
                              float* __restrict__ logits,
                              float* __restrict__ m) {
    int idx = blockIdx.x * blockDim.x + threadIdx.x;
    if (idx >= N_EDGES * NHEAD) return;
    int e  = idx >> 3;
    int hd = idx & 7;
    int r = (int)row[e], c = (int)col[e], t = (int)etype[e];
    float z = hl[r * NHEAD + hd] + hr[c * NHEAD + hd] + he[t * NHEAD + hd];
    z = (z > 0.0f) ? z : NEG_SLOPE * z;     // leaky_relu
    logits[idx] = z;
    atomic_max_f32(&m[c * NHEAD + hd], z);
}

// ------------------------------------------ 5) ex = exp(z - m[col]); denom
__global__ void exp_denom_kernel(const long long* __restrict__ col,
                                 const float* __restrict__ m,
                                 float* __restrict__ logits,   // in: z, out: ex
                                 float* __restrict__ denom) {
    int idx = blockIdx.x * blockDim.x + threadIdx.x;
    if (idx >= N_EDGES * NHEAD) return;
    int e  = idx >> 3;
    int hd = idx & 7;
    int c = (int)col[e];
    float ev = __expf(logits[idx] - m[c * NHEAD + hd]);
    logits[idx] = ev;
    atomicAdd(&denom[c * NHEAD + hd], ev);
}

// ------------------- 6) attn = ex/denom; scatter attn*h[row] into out[col]
__global__ void attn_scatter_kernel(const float* __restrict__ ex,
                                    const float* __restrict__ denom,
                                    const long long* __restrict__ row,
                                    const long long* __restrict__ col,
                                    const float* __restrict__ h,
                                    float* __restrict__ out,
                                    float* __restrict__ attn_out) {
    int idx = blockIdx.x * blockDim.x + threadIdx.x;
    if (idx >= N_EDGES * NHEAD) return;
    int e  = idx >> 3;
    int hd = idx & 7;
    int r = (int)row[e], c = (int)col[e];
    float a = ex[idx] / denom[c * NHEAD + hd];
    attn_out[idx] = a;
    const float4* hp = (const float4*)(h + (size_t)r * DHID + hd * D_OUT);
    float* op = out + (size_t)c * DHID + hd * D_OUT;
    #pragma unroll
    for (int q = 0; q < 8; ++q) {
        float4 v = hp[q];
        atomicAdd(op + 4 * q + 0, a * v.x);   // non-returning f32 adds -> L2
        atomicAdd(op + 4 * q + 1, a * v.y);
        atomicAdd(op + 4 * q + 2, a * v.z);
        atomicAdd(op + 4 * q + 3, a * v.w);
    }
}

// ---------------------------------------------------------------- launcher
extern "C" void kernel_launch(void* const* d_in, const int* in_sizes, int n_in,
                              void* d_out, int out_size, void* d_ws, size_t ws_size,
                              hipStream_t stream) {
    (void)in_sizes; (void)n_in; (void)out_size; (void)ws_size;

    const float*     x        = (const float*)d_in[0];
    const float*     W        = (const float*)d_in[1];
    const float*     W_e      = (const float*)d_in[2];
    const float*     edge_emb = (const float*)d_in[3];
    const float*     a_l      = (const float*)d_in[4];
    const float*     a_r      = (const float*)d_in[5];
    const float*     a_e      = (const float*)d_in[6];
    const long long* row      = (const long long*)d_in[7];
    const long long* col      = (const long long*)d_in[8];
    const long long* etype    = (const long long*)d_in[9];

    float* out      = (float*)d_out;                          // [N, 256]
    float* attn_out = out + (size_t)N_NODES * DHID;           // [E, 8]

    // workspace layout (floats)
    float* h      = (float*)d_ws;                             // N*256
    float* hl     = h  + (size_t)N_NODES * DHID;              // N*8
    float* hr     = hl + (size_t)N_NODES * NHEAD;             // N*8
    float* m      = hr + (size_t)N_NODES * NHEAD;             // N*8
    float* denom  = m  + (size_t)N_NODES * NHEAD;             // N*8
    float* he     = denom + (size_t)N_NODES * NHEAD;          // 64 (40 used)
    float* logits = he + 64;                                  // E*8

    const int TB = 256;
    const int n_out   = N_NODES * DHID;     // 12.8M
    const int n_nh    = N_NODES * NHEAD;    // 400k
    const int n_eh    = N_EDGES * NHEAD;    // 6.4M

    // init accumulators (d_out / d_ws are poisoned by harness)
    fill_kernel<<<(n_out + TB - 1) / TB, TB, 0, stream>>>(out, 0.0f, n_out);
    fill_kernel<<<(n_nh + TB - 1) / TB, TB, 0, stream>>>(m, -INFINITY, n_nh);
    fill_kernel<<<(n_nh + TB - 1) / TB, TB, 0, stream>>>(denom, 0.0f, n_nh);

    // 1) h = x @ W  (WMMA f32 16x16x4, W async-staged into 256 KB of LDS)
    gemm_h_kernel<<<N_NODES / 16, 256, (size_t)D_IN * DHID * sizeof(float),
                    stream>>>(x, W, h);

    // 2) per-node attention terms
    node_attn_kernel<<<(n_nh + TB - 1) / TB, TB, 0, stream>>>(h, a_l, a_r, hl, hr);

    // 3) per-etype attention terms
    etype_attn_kernel<<<N_ETYPES * NHEAD, 64, 0, stream>>>(edge_emb, W_e, a_e, he);

    // 4) logits + segment max
    logits_kernel<<<(n_eh + TB - 1) / TB, TB, 0, stream>>>(hl, hr, he, row, col,
                                                           etype, logits, m);

    // 5) exp + segment sum
    exp_denom_kernel<<<(n_eh + TB - 1) / TB, TB, 0, stream>>>(col, m, logits, denom);

    // 6) normalize + weighted scatter-aggregate
    attn_scatter_kernel<<<(n_eh + TB - 1) / TB, TB, 0, stream>>>(logits, denom, row,
                                                                 col, h, out, attn_out);
}